// CausalMultimodal_62500364091935
// MI455X (gfx1250) — compile-verified
//
#include <hip/hip_runtime.h>

typedef float v2f __attribute__((ext_vector_type(2)));
typedef float v8f __attribute__((ext_vector_type(8)));

#define ALPHA_THR 0.1f
#define DIM 32
#define HID 4

__global__ __launch_bounds__(256) void causal_mm_fused(
    const float* __restrict__ Z, const float* __restrict__ matrix,
    const float* __restrict__ W1, const float* __restrict__ b1,
    const float* __restrict__ W2, const float* __restrict__ b2,
    float* __restrict__ E, int B)
{
    const int lane          = threadIdx.x & 31;
    const int waveInBlock   = threadIdx.x >> 5;
    const int wavesPerBlock = blockDim.x >> 5;
    const int waveId        = blockIdx.x * wavesPerBlock + waveInBlock;
    const int totalWaves    = gridDim.x * wavesPerBlock;

    const int l16 = lane & 15;  // L % 16
    const int hi  = lane >> 4;  // L / 16

    // ---------------- per-wave setup (wave-invariant across iterations) ----
    // B-fragments of maskT for WMMA 16x16x4:
    //   lane L holds Bmat[K = 4k + 2*hi + {0,1}][N = nt*16 + l16]
    //   Bmat[K][N] = mask[N][K] = (matrix[N*32+K] > thr)
    v2f bfrag[2][8];
#pragma unroll
    for (int nt = 0; nt < 2; ++nt) {
        const int row = nt * 16 + l16;
#pragma unroll
        for (int k = 0; k < 8; ++k) {
            const int col = 4 * k + 2 * hi;
            const float m0 = matrix[row * DIM + col + 0];
            const float m1 = matrix[row * DIM + col + 1];
            bfrag[nt][k].x = (m0 > ALPHA_THR) ? 1.0f : 0.0f;
            bfrag[nt][k].y = (m1 > ALPHA_THR) ? 1.0f : 0.0f;
        }
    }

    // Per-lane dim constants for the two N-tiles (n = t*16 + l16).
    // invc is folded into the agg-path weights: w1bi = (1/count) * W1[:,1,:]
    // (count==0 -> invc=0 -> folded weights 0 -> agg contribution 0, matching ref).
    float w1a[2][HID], w1bi[2][HID], bb1[2][HID], ww2[2][HID], bb2[2];
#pragma unroll
    for (int t = 0; t < 2; ++t) {
        const int n = t * 16 + l16;
        int cnt = 0;
#pragma unroll
        for (int j = 0; j < DIM; ++j)
            cnt += (matrix[n * DIM + j] > ALPHA_THR) ? 1 : 0;
        const float invc = (cnt > 0) ? (1.0f / (float)cnt) : 0.0f;
#pragma unroll
        for (int h = 0; h < HID; ++h) {
            w1a[t][h]  = W1[(n * 2 + 0) * HID + h];
            w1bi[t][h] = invc * W1[(n * 2 + 1) * HID + h];
            bb1[t][h]  = b1[n * HID + h];
            ww2[t][h]  = W2[n * HID + h];
        }
        bb2[t] = b2[n];
    }

    // ---------------- main loop: 16 batch rows per iteration ---------------
    const int nchunk = B >> 4;
    for (int chunk = waveId; chunk < nchunk; chunk += totalWaves) {
        const long r0 = (long)chunk << 4;

        // Prefetch next chunk's Z tile for this wave.
        const long nxt = (long)(chunk + totalWaves);
        if (nxt < (long)nchunk)
            __builtin_prefetch(Z + (nxt << 4) * DIM + lane * 16, 0, 1);

        // A-fragment source: lane L reads row r0 + l16, cols 4k + 2*hi (+1).
        const float* zrow = Z + (r0 + l16) * DIM + 2 * hi;

        v8f acc0 = {};
        v8f acc1 = {};
#pragma unroll
        for (int k = 0; k < 8; ++k) {
            v2f a = *(const v2f*)(zrow + 4 * k);
            acc0 = __builtin_amdgcn_wmma_f32_16x16x4_f32(
                false, a, false, bfrag[0][k], (short)0, acc0, false, false);
            acc1 = __builtin_amdgcn_wmma_f32_16x16x4_f32(
                false, a, false, bfrag[1][k], (short)0, acc1, false, false);
        }

        // Epilogue: per-dim MLP on [z, agg]; C layout: VGPR v, lane L
        //   -> element (M = v + 8*hi, N = t*16 + l16)
#pragma unroll
        for (int t = 0; t < 2; ++t) {
            const v8f ps = t ? acc1 : acc0;
            const int n = t * 16 + l16;
#pragma unroll
            for (int v = 0; v < 8; ++v) {
                const long row = r0 + v + 8 * hi;
                const float z = Z[row * DIM + n];   // L0-resident (tile just read)
                float e = bb2[t];
#pragma unroll
                for (int h = 0; h < HID; ++h) {
                    float hd = fmaf(z, w1a[t][h], fmaf(ps[v], w1bi[t][h], bb1[t][h]));
                    hd = fmaxf(hd, 0.0f);
                    e = fmaf(hd, ww2[t][h], e);
                }
                __builtin_nontemporal_store(e, E + row * DIM + n);
            }
        }
    }
}

extern "C" void kernel_launch(void* const* d_in, const int* in_sizes, int n_in,
                              void* d_out, int out_size, void* d_ws, size_t ws_size,
                              hipStream_t stream) {
    const float* Z      = (const float*)d_in[0];
    const float* matrix = (const float*)d_in[1];
    const float* W1     = (const float*)d_in[2];
    const float* b1     = (const float*)d_in[3];
    const float* W2     = (const float*)d_in[4];
    const float* b2     = (const float*)d_in[5];
    float* E = (float*)d_out;

    const int B = in_sizes[0] / DIM;          // 1,048,576
    const int block = 256;                    // 8 waves / block
    const int wavesPerBlock = block / 32;
    const int nchunk = B >> 4;
    int blocks = (nchunk + wavesPerBlock - 1) / wavesPerBlock;
    if (blocks > 2048) blocks = 2048;         // persistent waves, grid-stride
    if (blocks < 1) blocks = 1;

    causal_mm_fused<<<blocks, block, 0, stream>>>(Z, matrix, W1, b1, W2, b2, E, B);
}